// CTUNOBlock1D_7172595384344
// MI455X (gfx1250) — compile-verified
//
#include <hip/hip_runtime.h>

// ---------------- problem constants ----------------
#define B_      32
#define L_      8192
#define CIN_    64
#define COUT_   64
#define TEMB_   256
#define M_      33          // retained rfft modes
#define KK_     66          // 2*M interleaved re/im rows
#define KKP_    68          // padded to multiple of 4 (synthesis K)
#define MROWS_  80          // analysis rows padded to multiple of 16
#define NSPLIT_ 8           // split-K factor for the analysis GEMM
#define KSLICE_ (L_ / NSPLIT_)
#define NBLK_C  (B_ * (L_ / 128))   // k_combine blocks = 2048
#define EPS_    1e-5f
#define NSTAT_  ((float)(B_ * L_))

typedef float v2f __attribute__((ext_vector_type(2)));
typedef float v4f __attribute__((ext_vector_type(4)));
typedef float v8f __attribute__((ext_vector_type(8)));

static __device__ __forceinline__ v8f wmma_f32(v2f a, v2f b, v8f c) {
    // V_WMMA_F32_16X16X4_F32 : D(16x16 f32) = A(16x4 f32) * B(4x16 f32) + C
    return __builtin_amdgcn_wmma_f32_16x16x4_f32(false, a, false, b, (short)0, c,
                                                 false, false);
}

// ---------------- workspace layout (float offsets) ----------------
#define WS_STATS  0                                   // 128 final (sum|sumsq)
#define WS_TCRE   128                                 // B*M
#define WS_TCIM   (WS_TCRE + B_ * M_)
#define WS_PSI    (WS_TCIM + B_ * M_)                 // B*128 (w_t | b_t)
#define WS_BF     (WS_PSI + B_ * 128)                 // MROWS_*L_ analysis basis
#define WS_A2     (WS_BF + MROWS_ * L_)               // L_*KKP_ synthesis basis
#define WS_XC     (WS_A2 + L_ * KKP_)                 // B*MROWS_*64 reduced spectra
#define WS_XCP    (WS_XC + B_ * MROWS_ * COUT_)       // NSPLIT_ partial spectra
#define WS_BC     (WS_XCP + NSPLIT_ * B_ * MROWS_ * COUT_) // B*KKP_*64 mixed modes
#define WS_PART   (WS_BC + B_ * KKP_ * COUT_)         // NBLK_C*128 block stats

// ---------------- kernel 1: DFT basis tables + zero Y ----------------
__global__ void k_tables(float* __restrict__ ws) {
    const float inv_sc = 1.0f / (float)L_;
    const float w0 = 6.283185307179586476925f / (float)L_;
    const int total = MROWS_ * L_ + L_ * KKP_ + B_ * KKP_ * COUT_;
    for (int t = blockIdx.x * blockDim.x + threadIdx.x; t < total;
         t += gridDim.x * blockDim.x) {
        if (t < MROWS_ * L_) {
            // analysis: Bf[2k][n] = cos(2pi k n/L)/L ; Bf[2k+1][n] = -sin(...)/L
            int row = t / L_, n = t - row * L_;
            float v = 0.f;
            if (row < KK_) {
                int k = row >> 1;
                int m = (k * n) & (L_ - 1);      // angle reduction mod L (L = 2^13)
                float ang = w0 * (float)m;
                v = (row & 1) ? (-sinf(ang) * inv_sc) : (cosf(ang) * inv_sc);
            }
            ws[WS_BF + t] = v;
        } else if (t < MROWS_ * L_ + L_ * KKP_) {
            // synthesis: A2[n][2k] = c_k cos ; A2[n][2k+1] = -c_k sin ; c_0=1 else 2
            int t2 = t - MROWS_ * L_;
            int n = t2 / KKP_, kk = t2 - n * KKP_;
            float v = 0.f;
            if (kk < KK_) {
                int k = kk >> 1;
                int m = (k * n) & (L_ - 1);
                float ang = w0 * (float)m;
                float c = (k == 0) ? 1.f : 2.f;
                v = (kk & 1) ? (-c * sinf(ang)) : (c * cosf(ang));
            }
            ws[WS_A2 + t2] = v;
        } else {
            ws[WS_BC + (t - MROWS_ * L_ - L_ * KKP_)] = 0.f;  // zero pad rows of Y
        }
    }
}

// ---------------- kernel 2: t_c, psi ----------------
__global__ void k_small(const float* __restrict__ t_emb,
                        const float* __restrict__ dense_re,
                        const float* __restrict__ dense_im,
                        const float* __restrict__ psi_kernel,
                        float* __restrict__ ws) {
    int tid = blockIdx.x * blockDim.x + threadIdx.x;
    if (tid < B_ * M_) {
        int b = tid / M_, k = tid - b * M_;
        float re = 0.f, im = 0.f;
        for (int t = 0; t < TEMB_; ++t) {
            float e = t_emb[b * TEMB_ + t];
            re += e * dense_re[t * M_ + k];
            im += e * dense_im[t * M_ + k];
        }
        ws[WS_TCRE + tid] = re;
        ws[WS_TCIM + tid] = im;
    }
    int p = tid - B_ * M_;
    if (p >= 0 && p < B_ * 128) {                       // psi = t_emb @ psi_kernel
        int b = p >> 7, c = p & 127;
        float s = 0.f;
        for (int t = 0; t < TEMB_; ++t)
            s += t_emb[b * TEMB_ + t] * psi_kernel[t * 128 + c];
        ws[WS_PSI + p] = s;
    }
}

// ---------------- kernel 3: split-K analysis GEMM (truncated rfft) ----------------
// XcP[s][b] (80x64) += Bf[:, sK:(s+1)K] @ x[b][sK:(s+1)K, :]; one tile per wave.
__global__ void __launch_bounds__(256) k_fwd(const float* __restrict__ x,
                                             float* __restrict__ ws) {
    const float* Bf = ws + WS_BF;
    float* XcP = ws + WS_XCP;
    int wave = (blockIdx.x << 3) + (threadIdx.x >> 5);  // 0..5119
    int lane = threadIdx.x & 31;
    int half = lane >> 4, idx = lane & 15;
    int s = wave / (B_ * 20);
    int rem = wave - s * (B_ * 20);
    int b = rem / 20;
    int rt = (rem % 20) >> 2;
    int ct = rem & 3;
    int k0 = s * KSLICE_;
    const float* Bfr = Bf + (size_t)(rt * 16 + idx) * L_ + k0;
    const float* xb = x + ((size_t)b * L_ + k0) * CIN_;
    int col = ct * 16 + idx;
    v8f acc = {};
    for (int k = 0; k < KSLICE_; k += 4) {
        int kb = k + 2 * half;
        int kpf = (k + 128 < KSLICE_) ? k + 128 : k;
        __builtin_prefetch(xb + (size_t)kpf * CIN_ + col, 0, 1);
        v2f a = *(const v2f*)(Bfr + kb);                // basis row, contiguous
        v2f bb;
        bb.x = xb[(size_t)kb * CIN_ + col];
        bb.y = xb[(size_t)(kb + 1) * CIN_ + col];
        acc = wmma_f32(a, bb, acc);
    }
#pragma unroll
    for (int r = 0; r < 8; ++r) {
        int row = rt * 16 + r + 8 * half;
        XcP[(((size_t)s * B_ + b) * MROWS_ + row) * COUT_ + col] = acc[r];
    }
}

// ---------------- kernel 3b: deterministic split-K reduction ----------------
__global__ void k_reduce(float* __restrict__ ws) {
    int t = blockIdx.x * blockDim.x + threadIdx.x;
    const int n = B_ * MROWS_ * COUT_;
    if (t >= n) return;
    const float* XcP = ws + WS_XCP;
    float acc = 0.f;
#pragma unroll
    for (int s = 0; s < NSPLIT_; ++s) acc += XcP[(size_t)s * n + t];
    ws[WS_XC + t] = acc;
}

// ---------------- kernel 4: complex mode mixing ----------------
// Y[b,k,:] = t_c[b,k] * (X[b,k,:] @ W[k]) -> interleaved re/im rows of Bc
__global__ void k_mix(const float* __restrict__ wre,
                      const float* __restrict__ wim,
                      float* __restrict__ ws) {
    int tid = blockIdx.x * blockDim.x + threadIdx.x;
    if (tid >= B_ * M_ * COUT_) return;
    int b = tid / (M_ * COUT_);
    int rem = tid - b * (M_ * COUT_);
    int k = rem / COUT_, o = rem - k * COUT_;
    const float* Xc = ws + WS_XC + (size_t)b * MROWS_ * COUT_;
    float sr = 0.f, si = 0.f;
    for (int c = 0; c < CIN_; ++c) {
        float xr = Xc[(2 * k) * COUT_ + c];
        float xi = Xc[(2 * k + 1) * COUT_ + c];
        float ar = wre[((size_t)k * CIN_ + c) * COUT_ + o];
        float ai = wim[((size_t)k * CIN_ + c) * COUT_ + o];
        sr += xr * ar - xi * ai;
        si += xr * ai + xi * ar;
    }
    float tr = ws[WS_TCRE + b * M_ + k], ti = ws[WS_TCIM + b * M_ + k];
    float* Bcb = ws + WS_BC + (size_t)b * KKP_ * COUT_;
    Bcb[(2 * k) * COUT_ + o]     = tr * sr - ti * si;
    Bcb[(2 * k + 1) * COUT_ + o] = tr * si + ti * sr;
}

// ---------------- kernel 5: fused synthesis + residual + block stats ----------------
__global__ void __launch_bounds__(256) k_combine(
    const float* __restrict__ x, const float* __restrict__ conv_kernel,
    const float* __restrict__ conv_bias, const float* __restrict__ tc_weights,
    float* __restrict__ out, float* __restrict__ ws) {
    __shared__ float lds_h[8 * 16 * COUT_];        // 32 KB: per-wave h_mod slabs
    __shared__ float lds_s[4][8][2][16];           // per-(cb,wave,half,idx) sums
    __shared__ float lds_s2[4][8][2][16];
    const float* A2 = ws + WS_A2;
    const float* psi = ws + WS_PSI;

    int tid = threadIdx.x;
    int wave = tid >> 5, lane = tid & 31;
    int half = lane >> 4, idx = lane & 15;
    int b = blockIdx.x >> 6;
    int chunk = blockIdx.x & 63;
    int r0 = chunk * 128 + wave * 16;
    const float* xb = x + (size_t)b * L_ * CIN_;
    const float* Bcb = ws + WS_BC + (size_t)b * KKP_ * COUT_;
    float* hslab = lds_h + wave * 16 * COUT_;

    // GEMM1: h(16x64) = x_rows(16x64) @ conv_kernel(64x64)
    v8f acc1[4] = {};
    for (int k = 0; k < CIN_; k += 4) {
        int kb = k + 2 * half;
        v2f a = *(const v2f*)(xb + (size_t)(r0 + idx) * CIN_ + kb);
#pragma unroll
        for (int cb = 0; cb < 4; ++cb) {
            int col = cb * 16 + idx;
            v2f bb;
            bb.x = conv_kernel[kb * COUT_ + col];
            bb.y = conv_kernel[(kb + 1) * COUT_ + col];
            acc1[cb] = wmma_f32(a, bb, acc1[cb]);
        }
    }
    // bias + per-batch diagonal modulation, stage to wave-private LDS slab
#pragma unroll
    for (int cb = 0; cb < 4; ++cb) {
        int col = cb * 16 + idx;
        float bias = conv_bias[col];
        float wt = psi[b * 128 + col];
#pragma unroll
        for (int r = 0; r < 8; ++r)
            hslab[(r + 8 * half) * COUT_ + col] = (acc1[cb][r] + bias) * wt;
    }

    // GEMM2: x_res(16x64) = h_mod(16x64) @ tc_weights^T
    v8f acc2[4] = {};
    for (int k = 0; k < COUT_; k += 4) {
        int kb = k + 2 * half;
        v2f a = *(const v2f*)(hslab + idx * COUT_ + kb);
#pragma unroll
        for (int cb = 0; cb < 4; ++cb) {
            int col = cb * 16 + idx;
            v2f bb = *(const v2f*)(tc_weights + col * COUT_ + kb);
            acc2[cb] = wmma_f32(a, bb, acc2[cb]);
        }
    }

    // GEMM3: x_spec(16x64) = A2_rows(16x68) @ Y[b](68x64)
    v8f acc3[4] = {};
    for (int k = 0; k < KKP_; k += 4) {
        int kb = k + 2 * half;
        v2f a = *(const v2f*)(A2 + (size_t)(r0 + idx) * KKP_ + kb);
#pragma unroll
        for (int cb = 0; cb < 4; ++cb) {
            int col = cb * 16 + idx;
            v2f bb;
            bb.x = Bcb[kb * COUT_ + col];
            bb.y = Bcb[(kb + 1) * COUT_ + col];
            acc3[cb] = wmma_f32(a, bb, acc3[cb]);
        }
    }

    // combine + write + deterministic per-block stats (lane owns one col per cb)
#pragma unroll
    for (int cb = 0; cb < 4; ++cb) {
        int col = cb * 16 + idx;
        float bt = psi[b * 128 + 64 + col];
        float s = 0.f, s2 = 0.f;
#pragma unroll
        for (int r = 0; r < 8; ++r) {
            int grow = r0 + r + 8 * half;
            float v = acc2[cb][r] + acc3[cb][r] + bt;
            out[((size_t)b * L_ + grow) * COUT_ + col] = v;
            s += v;
            s2 += v * v;
        }
        lds_s[cb][wave][half][idx] = s;       // conflict-free, no atomics
        lds_s2[cb][wave][half][idx] = s2;
    }
    __syncthreads();
    if (tid < COUT_) {                        // fixed-order column reduction
        int cb = tid >> 4, i = tid & 15;
        float s = 0.f, s2 = 0.f;
#pragma unroll
        for (int w = 0; w < 8; ++w)
#pragma unroll
            for (int h = 0; h < 2; ++h) {
                s += lds_s[cb][w][h][i];
                s2 += lds_s2[cb][w][h][i];
            }
        ws[WS_PART + (size_t)blockIdx.x * 128 + tid] = s;
        ws[WS_PART + (size_t)blockIdx.x * 128 + 64 + tid] = s2;
    }
}

// ---------------- kernel 5b: deterministic global stats reduction ----------------
__global__ void k_stats(float* __restrict__ ws) {
    int tid = threadIdx.x;                    // single block, 128 threads
    if (tid >= 128) return;
    const float* part = ws + WS_PART;
    float acc = 0.f;
    for (int blk = 0; blk < NBLK_C; ++blk) acc += part[(size_t)blk * 128 + tid];
    ws[WS_STATS + tid] = acc;
}

// ---------------- kernel 6: batchnorm + relu, float4 vectorized ----------------
__global__ void k_bn(float* __restrict__ out, const float* __restrict__ bn_scale,
                     const float* __restrict__ bn_bias,
                     const float* __restrict__ ws) {
    const float* st = ws + WS_STATS;
    const size_t total4 = (size_t)B_ * L_ * COUT_ / 4;
    for (size_t i = blockIdx.x * (size_t)blockDim.x + threadIdx.x; i < total4;
         i += (size_t)gridDim.x * blockDim.x) {
        v4f v = *((v4f*)out + i);
        int c0 = (int)((i * 4) & 63);         // 64 % 4 == 0 -> no wrap inside v4
        v4f r;
#pragma unroll
        for (int j = 0; j < 4; ++j) {
            int col = c0 + j;
            float mean = st[col] / NSTAT_;
            float var = st[64 + col] / NSTAT_ - mean * mean;
            float y = (v[j] - mean) * rsqrtf(var + EPS_) * bn_scale[col] + bn_bias[col];
            r[j] = fmaxf(y, 0.f);
        }
        *((v4f*)out + i) = r;
    }
}

// ---------------- launcher ----------------
extern "C" void kernel_launch(void* const* d_in, const int* in_sizes, int n_in,
                              void* d_out, int out_size, void* d_ws, size_t ws_size,
                              hipStream_t stream) {
    (void)in_sizes; (void)n_in; (void)out_size; (void)ws_size;
    const float* x     = (const float*)d_in[0];
    const float* t_emb = (const float*)d_in[1];
    const float* wre   = (const float*)d_in[2];
    const float* wim   = (const float*)d_in[3];
    const float* dre   = (const float*)d_in[4];
    const float* dim_  = (const float*)d_in[5];
    const float* ck    = (const float*)d_in[6];
    const float* cb    = (const float*)d_in[7];
    const float* tc    = (const float*)d_in[8];
    const float* pk    = (const float*)d_in[9];
    const float* bs    = (const float*)d_in[10];
    const float* bb    = (const float*)d_in[11];
    float* out = (float*)d_out;
    float* ws  = (float*)d_ws;

    k_tables<<<2048, 256, 0, stream>>>(ws);
    k_small<<<21, 256, 0, stream>>>(t_emb, dre, dim_, pk, ws);
    k_fwd<<<NSPLIT_ * 80, 256, 0, stream>>>(x, ws);          // 5120 waves
    k_reduce<<<(B_ * MROWS_ * COUT_ + 255) / 256, 256, 0, stream>>>(ws);
    k_mix<<<(B_ * M_ * COUT_ + 255) / 256, 256, 0, stream>>>(wre, wim, ws);
    k_combine<<<NBLK_C, 256, 0, stream>>>(x, ck, cb, tc, out, ws);
    k_stats<<<1, 128, 0, stream>>>(ws);
    k_bn<<<4096, 256, 0, stream>>>(out, bs, bb, ws);
}